// AnchorAugHead_71270687310618
// MI455X (gfx1250) — compile-verified
//
#include <hip/hip_runtime.h>
#include <cstdint>
#include <cstddef>
#include <cmath>

#define A_CNT   45600
#define NIMG    8
#define KSEL    1500
#define MTOT    (NIMG * KSEL)     // 12000
#define NT16    (MTOT / 16)       // 750 column tiles of 16
#define NTRI    ((NT16 * (NT16 + 1)) / 2)   // 281625 upper-tri tiles (incl. diagonal)
#define IOU_THR 0.5f
#define NMS_THR 0.7f

typedef __attribute__((ext_vector_type(2))) float v2f;
typedef __attribute__((ext_vector_type(8))) float v8f;

__device__ __forceinline__ float boxArea(float4 b) { return (b.z - b.x) * (b.w - b.y); }

// Order-preserving float -> uint key for atomic max over signed floats.
__device__ __forceinline__ unsigned fkey(float v) {
  unsigned f = __float_as_uint(v);
  return (f & 0x80000000u) ? ~f : (f | 0x80000000u);
}

__global__ void k_init(unsigned* maxKey) {
  if (threadIdx.x == 0) *maxKey = 0u;
}

// IoU of each of 8 GT boxes vs all 45600 anchors.
__global__ void k_overlaps(const float4* __restrict__ gt, const float4* __restrict__ prop,
                           float* __restrict__ ov) {
  int a = blockIdx.x * 256 + threadIdx.x;
  if (a >= A_CNT) return;
  float4 p = prop[a];
  float ap = boxArea(p);
  #pragma unroll
  for (int i = 0; i < NIMG; ++i) {
    float4 g = gt[i];
    float w = fminf(p.z, g.z) - fmaxf(p.x, g.x); w = fmaxf(w, 0.f);
    float h = fminf(p.w, g.w) - fmaxf(p.y, g.y); h = fmaxf(h, 0.f);
    float inter = w * h;
    float uni = ap + boxArea(g) - inter;
    ov[i * A_CNT + a] = inter / fmaxf(uni, 1e-6f);
  }
}

// Exact top-K by rank counting (stable ties by index), LDS-tiled O(A^2) per image.
// Next-chunk global_prefetch_b8 overlaps HBM/L2 latency with the LDS compare loop.
__global__ void k_topk(const float* __restrict__ ov, float* __restrict__ topv,
                       int* __restrict__ topi) {
  int img = blockIdx.y;
  int a = blockIdx.x * 256 + threadIdx.x;
  const float* v = ov + (size_t)img * A_CNT;
  float mine = (a < A_CNT) ? v[a] : -1e30f;
  __shared__ float sv[256];
  int rank = 0;
  for (int base = 0; base < A_CNT; base += 256) {
    int idx = base + threadIdx.x;
    sv[threadIdx.x] = (idx < A_CNT) ? v[idx] : -1e30f;
    if (base + 256 + (int)threadIdx.x < A_CNT)
      __builtin_prefetch(&v[base + 256 + threadIdx.x], 0, 0);   // global_prefetch_b8
    __syncthreads();
    int lim = min(256, A_CNT - base);
    for (int t = 0; t < lim; ++t) {
      float o = sv[t];
      int j = base + t;
      rank += (o > mine || (o == mine && j < a)) ? 1 : 0;
    }
    __syncthreads();
  }
  if (a < A_CNT && rank < KSEL) {
    topv[img * KSEL + rank] = mine;
    topi[img * KSEL + rank] = a;
  }
}

// Gather candidate boxes, build validity + scores, reduce global max coordinate.
__global__ void k_gather(const float4* __restrict__ prop, const float* __restrict__ topv,
                         const int* __restrict__ topi, const float* __restrict__ scores,
                         float4* __restrict__ bflat, float* __restrict__ sArr,
                         int* __restrict__ vvArr, unsigned* __restrict__ maxKey) {
  int m = blockIdx.x * 256 + threadIdx.x;
  float lmax = -3.0e38f;
  if (m < MTOT) {
    int img = m / KSEL;
    int a = topi[m];
    float4 b = prop[a];
    float tv = topv[m];
    bool anyPos = topv[img * KSEL] > IOU_THR;   // max IoU of this image
    bool valid = anyPos ? (tv > IOU_THR) : true;
    bflat[m] = b;
    vvArr[m] = valid ? 1 : 0;
    sArr[m]  = valid ? scores[m] : -1.0f;
    lmax = fmaxf(fmaxf(b.x, b.y), fmaxf(b.z, b.w));
  }
  __shared__ float red[256];
  red[threadIdx.x] = lmax;
  __syncthreads();
  for (int s = 128; s > 0; s >>= 1) {
    if (threadIdx.x < s) red[threadIdx.x] = fmaxf(red[threadIdx.x], red[threadIdx.x + s]);
    __syncthreads();
  }
  if (threadIdx.x == 0) atomicMax(maxKey, fkey(red[0]));
}

// Apply per-image coordinate offsets (batched_nms trick).
__global__ void k_offsets(const float4* __restrict__ bflat, const unsigned* __restrict__ maxKey,
                          float4* __restrict__ boff) {
  int m = blockIdx.x * 256 + threadIdx.x;
  if (m >= MTOT) return;
  unsigned k = *maxKey;
  float maxc = (k & 0x80000000u) ? __uint_as_float(k & 0x7FFFFFFFu) : __uint_as_float(~k);
  float off = (float)(m / KSEL) * (maxc + 1.0f);
  float4 b = bflat[m];
  b.x += off; b.y += off; b.z += off; b.w += off;
  boff[m] = b;
}

// Stable descending rank-sort by score; scatter sorted arrays directly.
__global__ void k_rank(const float* __restrict__ sArr, const int* __restrict__ vvArr,
                       const float4* __restrict__ bflat, const float4* __restrict__ boff,
                       float4* __restrict__ sFlat, float4* __restrict__ sOff,
                       float* __restrict__ sScore, int* __restrict__ sValid) {
  int m = blockIdx.x * 256 + threadIdx.x;
  float mine = (m < MTOT) ? sArr[m] : -2.0f;
  __shared__ float sv[256];
  int rank = 0;
  for (int base = 0; base < MTOT; base += 256) {
    int idx = base + threadIdx.x;
    sv[threadIdx.x] = (idx < MTOT) ? sArr[idx] : -3.0f;
    if (base + 256 + (int)threadIdx.x < MTOT)
      __builtin_prefetch(&sArr[base + 256 + threadIdx.x], 0, 0); // global_prefetch_b8
    __syncthreads();
    int lim = min(256, MTOT - base);
    for (int t = 0; t < lim; ++t) {
      float o = sv[t];
      int j = base + t;
      rank += (o > mine || (o == mine && j < m)) ? 1 : 0;
    }
    __syncthreads();
  }
  if (m < MTOT) {
    sFlat[rank]  = bflat[m];
    sOff[rank]   = boff[m];
    sScore[rank] = mine;
    sValid[rank] = vvArr[m];
  }
}

// One wave32 per 16x16 tile of the upper-triangular suppression matrix; 1-D
// triangular launch (no dead blocks). Union term area_i + area_j computed as a
// rank-2 product on the matrix unit:
//   A(16x4) = [area_i, 1, 0, 0],  B(4x16) = [1; area_j; 0; 0]
// D layout: row m = d + 8*(lane>=16), col n = lane&15  -> per-lane IoU + ballot.
__global__ void k_mask(const float4* __restrict__ boff, unsigned short* __restrict__ mask) {
  // decode linear tile index -> (rt, ct), ct >= rt; start(rt) = rt*N - rt*(rt-1)/2
  int idx = blockIdx.x;
  int rt = (int)((2.0 * NT16 + 1.0 -
                  sqrt((2.0 * NT16 + 1.0) * (2.0 * NT16 + 1.0) - 8.0 * (double)idx)) * 0.5);
  while (rt > 0 && rt * NT16 - (rt * (rt - 1)) / 2 > idx) --rt;
  while ((rt + 1) * NT16 - ((rt + 1) * rt) / 2 <= idx) ++rt;
  int ct = rt + (idx - (rt * NT16 - (rt * (rt - 1)) / 2));

  int lane = threadIdx.x;
  __shared__ float4 rb[16];
  __shared__ float4 cbs[16];
  if (lane < 16) {
    rb[lane]  = boff[rt * 16 + lane];
    cbs[lane] = boff[ct * 16 + lane];
  }
  __syncthreads();
  bool lo = lane < 16;
  float aR = lo ? boxArea(rb[lane & 15])  : 0.f;   // A: lanes 0-15 hold K=0,1; lanes 16-31 hold K=2,3 (=0)
  float aC = lo ? boxArea(cbs[lane & 15]) : 0.f;   // B: lanes 0-15 hold K=0,1 rows; K=2,3 rows zero
  v2f Am = { lo ? aR : 0.f, lo ? 1.f : 0.f };
  v2f Bm = { lo ? 1.f : 0.f, lo ? aC : 0.f };
  v8f Sacc = { 0.f, 0.f, 0.f, 0.f, 0.f, 0.f, 0.f, 0.f };
  // (neg_a, A, neg_b, B, c_mod, C, reuse_a, reuse_b) -> v_wmma_f32_16x16x4_f32
  Sacc = __builtin_amdgcn_wmma_f32_16x16x4_f32(false, Am, false, Bm, (short)0, Sacc,
                                               false, false);
  float4 c = cbs[lane & 15];
  int hi = lane >> 4;
  unsigned rmask = 0u;
  #pragma unroll
  for (int d = 0; d < 8; ++d) {
    float4 r = rb[d + 8 * hi];
    float w = fminf(r.z, c.z) - fmaxf(r.x, c.x); w = fmaxf(w, 0.f);
    float h = fminf(r.w, c.w) - fmaxf(r.y, c.y); h = fmaxf(h, 0.f);
    float inter = w * h;
    // iou > thr  <=>  inter > thr * (areaSum - inter), areaSum from WMMA
    bool pred = inter > NMS_THR * (Sacc[d] - inter);
    unsigned bal = (unsigned)__ballot(pred);
    if (lane == d)     rmask = bal & 0xffffu;          // row d    (bits from lanes 0-15)
    if (lane == d + 8) rmask = (bal >> 16) & 0xffffu;  // row d+8  (bits from lanes 16-31)
  }
  if (lane < 16) mask[(size_t)(rt * 16 + lane) * NT16 + ct] = (unsigned short)rmask;
}

// Sequential greedy scan, single wave32; 750-entry u16 removal bitset in LDS.
// OR loop starts at tile i>>4, so the unwritten lower triangle is never read.
__global__ void k_scan(const int* __restrict__ sValid, const unsigned short* __restrict__ mask,
                       int* __restrict__ keep) {
  int lane = threadIdx.x;
  __shared__ unsigned short removal[NT16];
  for (int t = lane; t < NT16; t += 32) removal[t] = 0;
  __syncthreads();
  for (int i = 0; i < MTOT; ++i) {
    unsigned w = removal[i >> 4];
    bool kept = (sValid[i] != 0) && !((w >> (i & 15)) & 1u);   // uniform across wave
    if (lane == 0) keep[i] = kept ? 1 : 0;
    if (kept) {
      const unsigned short* row = mask + (size_t)i * NT16;
      for (int t = (i >> 4) + lane; t < NT16; t += 32) removal[t] |= row[t];
      __syncthreads();
    }
  }
}

// Emit [M,5] dets, zeroing suppressed/invalid rows.
__global__ void k_out(const float4* __restrict__ sFlat, const float* __restrict__ sScore,
                      const int* __restrict__ keep, float* __restrict__ out) {
  int r = blockIdx.x * 256 + threadIdx.x;
  if (r >= MTOT) return;
  float msc = keep[r] ? 1.0f : 0.0f;
  float4 b = sFlat[r];
  out[r * 5 + 0] = b.x * msc;
  out[r * 5 + 1] = b.y * msc;
  out[r * 5 + 2] = b.z * msc;
  out[r * 5 + 3] = b.w * msc;
  out[r * 5 + 4] = sScore[r] * msc;
}

extern "C" void kernel_launch(void* const* d_in, const int* in_sizes, int n_in,
                              void* d_out, int out_size, void* d_ws, size_t ws_size,
                              hipStream_t stream) {
  const float4* gt     = (const float4*)d_in[0];   // [8,4]
  const float4* prop   = (const float4*)d_in[1];   // [45600,4]
  const float*  scores = (const float*)d_in[2];    // [8,1500]
  float* out = (float*)d_out;                      // [12000,5]

  uint8_t* p = (uint8_t*)d_ws;
  auto take = [&](size_t bytes) -> void* {
    void* r = (void*)p;
    p += (bytes + 15) & ~(size_t)15;
    return r;
  };
  float*          ov      = (float*)take((size_t)NIMG * A_CNT * 4);     // 1.46 MB
  float*          topv    = (float*)take((size_t)MTOT * 4);
  int*            topi    = (int*)take((size_t)MTOT * 4);
  float4*         bflat   = (float4*)take((size_t)MTOT * 16);
  float4*         boff    = (float4*)take((size_t)MTOT * 16);
  float*          sArr    = (float*)take((size_t)MTOT * 4);
  int*            vvArr   = (int*)take((size_t)MTOT * 4);
  float4*         sFlat   = (float4*)take((size_t)MTOT * 16);
  float4*         sOff    = (float4*)take((size_t)MTOT * 16);
  float*          sScore  = (float*)take((size_t)MTOT * 4);
  int*            sValid  = (int*)take((size_t)MTOT * 4);
  unsigned*       maxKey  = (unsigned*)take(16);
  unsigned short* mask    = (unsigned short*)take((size_t)MTOT * NT16 * 2);  // 18 MB
  int*            keep    = (int*)take((size_t)MTOT * 4);
  (void)in_sizes; (void)n_in; (void)out_size; (void)ws_size;

  k_init<<<1, 32, 0, stream>>>(maxKey);
  k_overlaps<<<(A_CNT + 255) / 256, 256, 0, stream>>>(gt, prop, ov);
  k_topk<<<dim3((A_CNT + 255) / 256, NIMG), 256, 0, stream>>>(ov, topv, topi);
  k_gather<<<(MTOT + 255) / 256, 256, 0, stream>>>(prop, topv, topi, scores,
                                                   bflat, sArr, vvArr, maxKey);
  k_offsets<<<(MTOT + 255) / 256, 256, 0, stream>>>(bflat, maxKey, boff);
  k_rank<<<(MTOT + 255) / 256, 256, 0, stream>>>(sArr, vvArr, bflat, boff,
                                                 sFlat, sOff, sScore, sValid);
  k_mask<<<NTRI, 32, 0, stream>>>(sOff, mask);
  k_scan<<<1, 32, 0, stream>>>(sValid, mask, keep);
  k_out<<<(MTOT + 255) / 256, 256, 0, stream>>>(sFlat, sScore, keep, out);
}